// AdaptiveChebConv_57200374448667
// MI455X (gfx1250) — compile-verified
//
#include <hip/hip_runtime.h>
#include <hip/hip_bf16.h>

// Problem constants (compile-time, from the reference)
#define Bn 16
#define Nn 1024
#define Fn 64
#define Tn 12
#define On 64

typedef __attribute__((ext_vector_type(2))) float v2f;
typedef __attribute__((ext_vector_type(8))) float v8f;
typedef __attribute__((ext_vector_type(4))) unsigned int u32x4;
typedef __attribute__((ext_vector_type(4))) int i32x4;
typedef __attribute__((ext_vector_type(8))) int i32x8;

// D = A(16x4 f32) * B(4x16 f32) + C(16x16 f32), wave32, fp32 matrix pipe.
__device__ __forceinline__ v8f wmma4(v2f a, v2f b, v8f c) {
  return __builtin_amdgcn_wmma_f32_16x16x4_f32(false, a, false, b, (short)0, c,
                                               false, false);
}

// Low 32 bits of a flat shared-pointer == workgroup-relative LDS byte offset.
__device__ __forceinline__ unsigned lds_off(const void* p) {
  return (unsigned)(unsigned long long)p;
}

// Tensor Data Mover: DMA a 2D f32 tile (tileRows x tileCols, row stride
// rowStride elems) from global memory into LDS, padding each row of
// 2^(padIntCode+1) DWORDs with 4 DWORDs so it lands in a [rows][cols+4]
// bank-conflict-free layout. Tracked by TENSORcnt. Issue once per workgroup
// (TDM ignores EXEC; call from one wave only).
__device__ __forceinline__ void tdm_load_tile(const float* gptr,
                                              unsigned ldsOff, int tileCols,
                                              int tileRows, int rowStride,
                                              int padIntCode) {
  unsigned long long ga = (unsigned long long)gptr;
  u32x4 g0;
  g0.x = 1u;                                    // count=1, user descriptor
  g0.y = ldsOff;                                // lds_addr
  g0.z = (unsigned)ga;                          // global_addr[31:0]
  g0.w = (unsigned)((ga >> 32) & 0x01FFFFFFull) // global_addr[56:32]
         | (2u << 30);                          // type=2 ("image")
  i32x8 g1;
  // workgroup_mask=0 | data_size=4B | pad_enable | pad_interval | pad_amount=4
  g1.s0 = (int)(0x00020000u | (1u << 20) | ((unsigned)padIntCode << 22) |
                (3u << 25));
  g1.s1 = 0;                      // atomic_barrier=0 | tensor_dim0(=1<<20) lo16
  g1.s2 = 16;                     // tensor_dim0 hi16 | tensor_dim1(=1<<20) lo16
  g1.s3 = 16 | (tileCols << 16);  // tensor_dim1 hi16 | tile_dim0
  g1.s4 = tileRows;               // tile_dim1 | tile_dim2=0
  g1.s5 = rowStride;              // tensor_dim0_stride lo32
  g1.s6 = 0;
  g1.s7 = 0;
  i32x4 z4 = {0, 0, 0, 0};
#if __clang_major__ >= 23
  i32x8 z8 = {0, 0, 0, 0, 0, 0, 0, 0};
  __builtin_amdgcn_tensor_load_to_lds(g0, g1, z4, z4, z8, 0);
#else
  __builtin_amdgcn_tensor_load_to_lds(g0, g1, z4, z4, 0);
#endif
}

// ---------------------------------------------------------------------------
// Kernel 1: adj_b[b,i,j] = adj[i,j] * spatial_attention[b,i,j]  (float4)
// ---------------------------------------------------------------------------
__global__ void __launch_bounds__(256)
adjb_mul_kernel(const float* __restrict__ adj, const float* __restrict__ sa,
                float* __restrict__ adjb) {
  int i = blockIdx.x * 256 + threadIdx.x;               // float4 index
  const int nn4 = (Nn * Nn) / 4;                        // power of two
  float4 a = ((const float4*)adj)[i & (nn4 - 1)];
  float4 s = ((const float4*)sa)[i];
  ((float4*)adjb)[i] = make_float4(a.x * s.x, a.y * s.y, a.z * s.z, a.w * s.w);
}

// ---------------------------------------------------------------------------
// Kernel 2: S2[b] = adj_b[b] @ adj_b[b]   (batched 1024x1024x1024, f32 WMMA)
// WG = 256 threads = 8 waves; WG tile 64(M) x 32(N); wave tile 16x16.
// Tiles staged by the Tensor Data Mover, double-buffered in LDS.
// ---------------------------------------------------------------------------
__global__ void __launch_bounds__(256)
s2_gemm_kernel(const float* __restrict__ adjb, float* __restrict__ s2) {
  __shared__ __align__(16) float As[2][64][36];  // [buf][m][k], TDM-padded
  __shared__ __align__(16) float Bs[2][32][36];  // [buf][k][n], TDM-padded

  const int b = blockIdx.z;
  const int m0 = blockIdx.x * 64;
  const int n0 = blockIdx.y * 32;
  const int tid = threadIdx.x;
  const int lane = tid & 31, wid = tid >> 5;
  const int wm = (wid & 3) * 16;        // 4 waves along M
  const int wn = (wid >> 2) * 16;       // 2 waves along N
  const int laneM = lane & 15;
  const int kB = (lane >> 4) * 2;       // lanes 16-31 hold K+2,K+3

  const float* Ab = adjb + (size_t)b * Nn * Nn;
  v8f c = {};

  // Prologue: DMA first K-chunk into buffer 0 (wave 0 issues for the WG).
  if (wid == 0) {
    tdm_load_tile(Ab + (size_t)m0 * Nn, lds_off(&As[0][0][0]), 32, 64, Nn, 4);
    tdm_load_tile(Ab + n0, lds_off(&Bs[0][0][0]), 32, 32, Nn, 4);
  }

  int buf = 0;
  for (int kc = 0; kc < Nn; kc += 32, buf ^= 1) {
    if (wid == 0) {
      if (kc + 32 < Nn) {
        // Issue next chunk into the other buffer, then wait until only those
        // two TDM ops remain outstanding (TDM is in-order per wave, so the
        // current buffer is complete).
        tdm_load_tile(Ab + (size_t)m0 * Nn + kc + 32,
                      lds_off(&As[buf ^ 1][0][0]), 32, 64, Nn, 4);
        tdm_load_tile(Ab + (size_t)(kc + 32) * Nn + n0,
                      lds_off(&Bs[buf ^ 1][0][0]), 32, 32, Nn, 4);
        __builtin_amdgcn_s_wait_tensorcnt(2);
      } else {
        __builtin_amdgcn_s_wait_tensorcnt(0);
      }
    }
    __syncthreads();  // current buffer visible to all waves
#pragma unroll
    for (int kk = 0; kk < 32; kk += 4) {
      v2f a = *(const v2f*)&As[buf][wm + laneM][kk + kB];  // M=laneM, K=kB..
      v2f bb;
      bb.x = Bs[buf][kk + kB][wn + laneM];                 // K rows / VGPR
      bb.y = Bs[buf][kk + kB + 1][wn + laneM];
      c = wmma4(a, bb, c);
    }
    __syncthreads();  // all reads of buf done before TDM overwrites it
  }

  float* Cb = s2 + (size_t)b * Nn * Nn;
  const int row0 = m0 + wm + (lane >> 4) * 8;   // VGPR j -> M = j (+8 hi half)
  const int col = n0 + wn + laneM;
#pragma unroll
  for (int j = 0; j < 8; ++j) Cb[(size_t)(row0 + j) * Nn + col] = c[j];
}

// ---------------------------------------------------------------------------
// Kernel 3: W[k-1][b,t][m][o] = sum_f x[b,m,f,t] * Theta[k,f,o]   (k=1,2)
// Per (b,t,k): GEMM M=1024(m) K=64(f) N=64(o). WG tile 64x64, wave 16x32.
// ---------------------------------------------------------------------------
__global__ void __launch_bounds__(256)
xw_gemm_kernel(const float* __restrict__ x, const float* __restrict__ Theta,
               float* __restrict__ W) {
  __shared__ __align__(16) float Xs[64][Fn + 4];  // [m][f]
  __shared__ __align__(16) float Ts[Fn][On + 4];  // [f][o]

  const int m0 = blockIdx.x * 64;
  const int bt = blockIdx.y;                    // b*T + t
  const int b = bt / Tn, t = bt % Tn;
  const int k = blockIdx.z + 1;                 // 1 or 2
  const int tid = threadIdx.x;

  // x tile: strided scalar loads (stride T between f)
#pragma unroll
  for (int i = 0; i < 16; ++i) {
    int lin = i * 256 + tid;
    int r = lin >> 6, f = lin & 63;
    const float* gp = &x[((size_t)(b * Nn + m0 + r) * Fn + f) * Tn + t];
    Xs[r][f] = *gp;
  }
  // Theta[k] tile: contiguous float4
#pragma unroll
  for (int i = 0; i < 4; ++i) {
    int lin = i * 256 + tid;
    int r = lin >> 4, c4 = (lin & 15) * 4;
    *(float4*)&Ts[r][c4] =
        *(const float4*)(Theta + (size_t)k * Fn * On + (size_t)r * On + c4);
  }
  __syncthreads();

  const int lane = tid & 31, wid = tid >> 5;
  const int wm = (wid & 3) * 16, wn = (wid >> 2) * 32;
  const int laneM = lane & 15, kB = (lane >> 4) * 2;
  v8f c0 = {}, c1 = {};
#pragma unroll
  for (int kk = 0; kk < Fn; kk += 4) {
    v2f a = *(const v2f*)&Xs[wm + laneM][kk + kB];
    v2f b0, b1;
    b0.x = Ts[kk + kB][wn + laneM];      b0.y = Ts[kk + kB + 1][wn + laneM];
    b1.x = Ts[kk + kB][wn + 16 + laneM]; b1.y = Ts[kk + kB + 1][wn + 16 + laneM];
    c0 = wmma4(a, b0, c0);
    c1 = wmma4(a, b1, c1);
  }

  float* Wp = W + ((size_t)(k - 1) * (Bn * Tn) + bt) * Nn * On;
  const int row0 = m0 + wm + (lane >> 4) * 8;
  const int col = wn + laneM;
#pragma unroll
  for (int j = 0; j < 8; ++j) {
    Wp[(size_t)(row0 + j) * On + col] = c0[j];
    Wp[(size_t)(row0 + j) * On + col + 16] = c1[j];
  }
}

// ---------------------------------------------------------------------------
// Kernel 4: out[b,n,o,t] = relu( sum_{k=1,2} sum_m S_k[b,m,n]*W_k[b,t,m,o]
//                                + sum_f x[b,n,f,t]*Theta[0,f,o] )
// Per (b,t): WG tile 64(n) x 64(o); K loop m=0..1023 (k=1,2) then f=0..63.
// B tiles (contiguous rows) staged by the TDM; transposed S tile manual.
// ---------------------------------------------------------------------------
__global__ void __launch_bounds__(256)
out_gemm_kernel(const float* __restrict__ x, const float* __restrict__ Theta,
                const float* __restrict__ adjb, const float* __restrict__ s2,
                const float* __restrict__ W, float* __restrict__ out) {
  __shared__ __align__(16) float As[64][36];  // [n][m-chunk] (transposed S)
  __shared__ __align__(16) float Bs[32][68];  // [m][o], TDM-padded (64+4)

  const int n0 = blockIdx.x * 64;
  const int bt = blockIdx.y;
  const int b = bt / Tn, t = bt % Tn;
  const int tid = threadIdx.x, lane = tid & 31, wid = tid >> 5;
  const int wm = (wid & 3) * 16, wn = (wid >> 2) * 32;
  const int laneM = lane & 15, kB = (lane >> 4) * 2;
  v8f c0 = {}, c1 = {};

  for (int k = 1; k <= 2; ++k) {
    const float* S = (k == 1 ? adjb : s2) + (size_t)b * Nn * Nn;
    const float* Wk = W + ((size_t)(k - 1) * (Bn * Tn) + bt) * Nn * On;
    for (int kc = 0; kc < Nn; kc += 32) {
      // Bs[m][o] = Wk[kc+m][o] via TDM (32 rows x 64 cols, pad 64->68)
      if (wid == 0)
        tdm_load_tile(Wk + (size_t)kc * On, lds_off(&Bs[0][0]), 64, 32, On, 5);
      // As[n][m] = S[kc+m][n0+n]: read rows contiguously, write transposed
#pragma unroll
      for (int i = 0; i < 2; ++i) {
        int lin = i * 256 + tid;
        int mr = lin >> 4, n4 = (lin & 15) * 4;
        float4 v = *(const float4*)(S + (size_t)(kc + mr) * Nn + n0 + n4);
        As[n4 + 0][mr] = v.x; As[n4 + 1][mr] = v.y;
        As[n4 + 2][mr] = v.z; As[n4 + 3][mr] = v.w;
      }
      if (wid == 0) __builtin_amdgcn_s_wait_tensorcnt(0);
      __syncthreads();
#pragma unroll
      for (int kk = 0; kk < 32; kk += 4) {
        v2f a = *(const v2f*)&As[wm + laneM][kk + kB];
        v2f b0, b1;
        b0.x = Bs[kk + kB][wn + laneM];      b0.y = Bs[kk + kB + 1][wn + laneM];
        b1.x = Bs[kk + kB][wn + 16 + laneM]; b1.y = Bs[kk + kB + 1][wn + 16 + laneM];
        c0 = wmma4(a, b0, c0);
        c1 = wmma4(a, b1, c1);
      }
      __syncthreads();
    }
  }

  // k=0 identity term: A[n][f] = x[b,n0+n,f,t], B = Theta[0] (TDM)
  for (int fc = 0; fc < Fn; fc += 32) {
    if (wid == 0)
      tdm_load_tile(Theta + (size_t)fc * On, lds_off(&Bs[0][0]), 64, 32, On, 5);
#pragma unroll
    for (int i = 0; i < 8; ++i) {
      int lin = i * 256 + tid;
      int r = lin >> 5, ff = lin & 31;
      As[r][ff] = x[((size_t)(b * Nn + n0 + r) * Fn + fc + ff) * Tn + t];
    }
    if (wid == 0) __builtin_amdgcn_s_wait_tensorcnt(0);
    __syncthreads();
#pragma unroll
    for (int kk = 0; kk < 32; kk += 4) {
      v2f a = *(const v2f*)&As[wm + laneM][kk + kB];
      v2f b0, b1;
      b0.x = Bs[kk + kB][wn + laneM];      b0.y = Bs[kk + kB + 1][wn + laneM];
      b1.x = Bs[kk + kB][wn + 16 + laneM]; b1.y = Bs[kk + kB + 1][wn + 16 + laneM];
      c0 = wmma4(a, b0, c0);
      c1 = wmma4(a, b1, c1);
    }
    __syncthreads();
  }

  // Epilogue: relu + strided store out[((b*N+n)*O+o)*T + t]
  const int row0 = n0 + wm + (lane >> 4) * 8;
  const int col = wn + laneM;
#pragma unroll
  for (int j = 0; j < 8; ++j) {
    float v0 = c0[j] > 0.f ? c0[j] : 0.f;
    float v1 = c1[j] > 0.f ? c1[j] : 0.f;
    out[((size_t)(b * Nn + row0 + j) * On + col) * Tn + t] = v0;
    out[((size_t)(b * Nn + row0 + j) * On + col + 16) * Tn + t] = v1;
  }
}

// ---------------------------------------------------------------------------
extern "C" void kernel_launch(void* const* d_in, const int* in_sizes, int n_in,
                              void* d_out, int out_size, void* d_ws,
                              size_t ws_size, hipStream_t stream) {
  (void)in_sizes; (void)n_in; (void)out_size; (void)ws_size;
  const float* x     = (const float*)d_in[0];  // [B,N,F,T]
  const float* sa    = (const float*)d_in[1];  // [B,N,N]
  const float* adj   = (const float*)d_in[2];  // [N,N]
  const float* Theta = (const float*)d_in[3];  // [K,F,O]
  float* out = (float*)d_out;                  // [B,N,O,T]

  // Workspace layout (floats): adj_b | S2 | W[2][B*T][N][O]
  float* ws   = (float*)d_ws;
  float* adjb = ws;
  float* s2   = ws + (size_t)Bn * Nn * Nn;
  float* W    = ws + 2 * (size_t)Bn * Nn * Nn;

  adjb_mul_kernel<<<(Bn * Nn * Nn / 4) / 256, 256, 0, stream>>>(adj, sa, adjb);
  s2_gemm_kernel<<<dim3(Nn / 64, Nn / 32, Bn), 256, 0, stream>>>(adjb, s2);
  xw_gemm_kernel<<<dim3(Nn / 64, Bn * Tn, 2), 256, 0, stream>>>(x, Theta, W);
  out_gemm_kernel<<<dim3(Nn / 64, Bn * Tn), 256, 0, stream>>>(x, Theta, adjb,
                                                              s2, W, out);
}